// UnsupervisedRGCN_36369783063047
// MI455X (gfx1250) — compile-verified
//
#include <hip/hip_runtime.h>
#include <hip/hip_bf16.h>

typedef __attribute__((ext_vector_type(2))) float v2f;
typedef __attribute__((ext_vector_type(8))) float v8f;

#define N_NODES 100000
#define R_REL   8
#define D_IN    128
#define EMB     128
#define E_EDGES 1600000
#define B_TRI   16384
#define MTILES  (N_NODES / 16)   // 6250, exact

#define WT_STRIDE 130            // transposed-W LDS row stride (floats), 8B-aligned rows
#define AT_STRIDE 132            // staged-A LDS row stride (floats), 16B-aligned rows
#define WT_FLOATS (EMB * WT_STRIDE)          // 16640 floats = 66,560 B
#define AT_FLOATS (8 * 16 * AT_STRIDE)       // 16896 floats = 67,584 B

// ---- workspace layout ----
#define WS_CNT_U    0
#define WS_HIST_U   800000
#define WS_CURSOR_U 800008
#define WS_OFFS_U   800016
#define WS_ZERO_U   800048
#define WS_PERM_B   3200128
#define WS_AGG_B    9600256

static __device__ inline v8f wmma4(v2f a, v2f b, v8f c) {
  return __builtin_amdgcn_wmma_f32_16x16x4_f32(
      false, a, false, b, (short)0, c, false, false);
}

// ---------------- setup passes ----------------

__global__ void k_zero(unsigned int* ws) {
  unsigned int i = blockIdx.x * blockDim.x + threadIdx.x;
  if (i < WS_ZERO_U) ws[i] = 0u;
}

__global__ void k_count(const int* __restrict__ erel, const int* __restrict__ edst,
                        unsigned int* __restrict__ cnt, unsigned int* __restrict__ hist) {
  int i = blockIdx.x * blockDim.x + threadIdx.x;
  if (i < E_EDGES) {
    int r = erel[i], d = edst[i];
    atomicAdd(&cnt[d * R_REL + r], 1u);
    atomicAdd(&hist[r], 1u);
  }
}

__global__ void k_scan(const unsigned int* __restrict__ hist,
                       unsigned int* __restrict__ offs,
                       unsigned int* __restrict__ cursor) {
  unsigned int acc = 0;
  for (int r = 0; r < R_REL; ++r) {
    offs[r] = acc;
    cursor[r] = acc;
    acc += hist[r];
  }
  offs[R_REL] = acc;
}

__global__ void k_bucket(const int* __restrict__ erel,
                         unsigned int* __restrict__ cursor,
                         int* __restrict__ perm) {
  int i = blockIdx.x * blockDim.x + threadIdx.x;
  if (i < E_EDGES) {
    int r = erel[i];
    unsigned int pos = atomicAdd(&cursor[r], 1u);
    perm[pos] = i;
  }
}

// ---------------- self-loop GEMM: agg = X @ W_self (initializes agg) ----------------

__global__ __launch_bounds__(256) void k_self_gemm(const float* __restrict__ X,
                                                   const float* __restrict__ Wself,
                                                   float* __restrict__ agg) {
  __shared__ float wt[WT_FLOATS];   // W_self transposed: wt[n*WT_STRIDE + k]
  for (int idx = threadIdx.x; idx < D_IN * EMB; idx += 256) {
    int k = idx >> 7, n = idx & 127;
    wt[n * WT_STRIDE + k] = Wself[idx];
  }
  __syncthreads();

  const int wave = threadIdx.x >> 5;
  const int lane = threadIdx.x & 31;
  const int mt = blockIdx.x * 8 + wave;
  if (mt >= MTILES) return;

  const int n    = lane & 15;
  const int half = lane >> 4;
  const float* xrow = X + (size_t)(mt * 16 + n) * D_IN;

  v8f acc[8] = {};
  for (int k = 0; k < D_IN; k += 4) {
    v2f a = *(const v2f*)&xrow[k + 2 * half];         // contiguous K pair
    #pragma unroll
    for (int nt = 0; nt < 8; ++nt) {
      v2f b = *(const v2f*)&wt[(nt * 16 + n) * WT_STRIDE + k + 2 * half];
      acc[nt] = wmma4(a, b, acc[nt]);
    }
  }
  #pragma unroll
  for (int nt = 0; nt < 8; ++nt) {
    #pragma unroll
    for (int v = 0; v < 8; ++v) {
      int row = mt * 16 + v + 8 * half;
      agg[(size_t)row * EMB + nt * 16 + n] = acc[nt][v];
    }
  }
}

// ------- fused per-relation edge GEMM + normalized scatter-add into agg -------

__global__ __launch_bounds__(256) void k_edge_gemm(const float* __restrict__ X,
                                                   const float* __restrict__ Wrel,
                                                   const int* __restrict__ esrc,
                                                   const int* __restrict__ edst,
                                                   const int* __restrict__ perm,
                                                   const unsigned int* __restrict__ offs,
                                                   const unsigned int* __restrict__ cnt,
                                                   float* __restrict__ agg) {
  const int rel = blockIdx.y;
  const unsigned int base  = offs[rel];
  const unsigned int count = offs[rel + 1] - base;
  if ((unsigned int)blockIdx.x * 128u >= count) return;

  // single shared block so LDS byte offsets for async copies are well-defined
  __shared__ float smem[WT_FLOATS + AT_FLOATS];   // ~131 KB of the 320 KB WGP LDS
  float* wt    = smem;                             // transposed W_rel[rel]
  float* atile = smem + WT_FLOATS;                 // staged gathered X rows (8 waves)

  const float* W = Wrel + (size_t)rel * D_IN * EMB;
  for (int idx = threadIdx.x; idx < D_IN * EMB; idx += 256) {
    int k = idx >> 7, n = idx & 127;
    wt[n * WT_STRIDE + k] = W[idx];
  }
  __syncthreads();

  const int wave = threadIdx.x >> 5;
  const int lane = threadIdx.x & 31;
  const unsigned int t0 = (unsigned int)blockIdx.x * 128u + (unsigned int)wave * 16u;
  if (t0 >= count) return;

  const int n    = lane & 15;
  const int half = lane >> 4;

  // ---- async-stage this wave's 16 gathered X rows into LDS (full 512B bursts) ----
  // LDS byte base of this wave's A tile (smem assumed at LDS offset 0).
  const unsigned atByteBase = (unsigned)(WT_FLOATS * 4) +
                              (unsigned)(wave * 16 * AT_STRIDE * 4);
  #pragma unroll
  for (int j = 0; j < 16; ++j) {
    unsigned int ej = t0 + (unsigned int)j;
    int e = perm[base + (ej < count ? ej : count - 1u)];
    unsigned srow = (unsigned)esrc[e];
    unsigned goff   = srow * (unsigned)(D_IN * 4) + (unsigned)lane * 16u; // byte off vs X
    unsigned ldsoff = atByteBase + (unsigned)(j * AT_STRIDE * 4) + (unsigned)lane * 16u;
    asm volatile("global_load_async_to_lds_b128 %0, %1, %2"
                 :
                 : "v"(ldsoff), "v"(goff), "s"(X)
                 : "memory");
  }
  asm volatile("s_wait_asynccnt 0x0" ::: "memory");

  const float* arow0 = atile + (size_t)wave * 16 * AT_STRIDE;

  v8f acc[8] = {};
  for (int k = 0; k < D_IN; k += 4) {
    // A frag: lane row m = lane&15, contiguous K pair from staged LDS tile
    v2f a = *(const v2f*)&arow0[n * AT_STRIDE + k + 2 * half];
    #pragma unroll
    for (int nt = 0; nt < 8; ++nt) {
      v2f b = *(const v2f*)&wt[(nt * 16 + n) * WT_STRIDE + k + 2 * half];
      acc[nt] = wmma4(a, b, acc[nt]);
    }
  }

  // scatter: lane holds column n; accumulator element v covers row m = v + 8*half
  #pragma unroll
  for (int v = 0; v < 8; ++v) {
    unsigned int eim = t0 + (unsigned int)(v + 8 * half);
    if (eim >= count) continue;
    int ee   = perm[base + eim];
    int drow = edst[ee];
    unsigned int c = cnt[drow * R_REL + rel];
    float norm = 1.0f / (float)(c > 0u ? c : 1u);
    float* outrow = agg + (size_t)drow * EMB;
    #pragma unroll
    for (int nt = 0; nt < 8; ++nt) {
      atomicAdd(&outrow[nt * 16 + n], norm * acc[nt][v]);
    }
  }
}

// ---------------- bias + ReLU (in place: agg -> H) ----------------

__global__ void k_bias_relu(const float* __restrict__ bvec, float* __restrict__ agg) {
  size_t i = (size_t)blockIdx.x * blockDim.x + threadIdx.x;
  if (i < (size_t)N_NODES * EMB) {
    float v = agg[i] + bvec[i & (EMB - 1)];
    agg[i] = v > 0.0f ? v : 0.0f;
  }
}

// ---------------- DistMult scoring: one wave per triple ----------------

__global__ __launch_bounds__(256) void k_score(const float* __restrict__ H,
                                               const float* __restrict__ rel_emb,
                                               const int* __restrict__ tri_nodes,
                                               const int* __restrict__ tri_rel,
                                               float* __restrict__ out) {
  int gt   = blockIdx.x * blockDim.x + threadIdx.x;
  int tri  = gt >> 5;
  int lane = threadIdx.x & 31;
  if (tri >= B_TRI) return;
  int s = tri_nodes[tri * 2 + 0];
  int o = tri_nodes[tri * 2 + 1];
  int r = tri_rel[tri];
  const float4* hs = (const float4*)(H + (size_t)s * EMB);
  const float4* ho = (const float4*)(H + (size_t)o * EMB);
  const float4* re = (const float4*)(rel_emb + (size_t)r * EMB);
  float4 a = hs[lane], b = ho[lane], c = re[lane];
  float sum = a.x * b.x * c.x + a.y * b.y * c.y + a.z * b.z * c.z + a.w * b.w * c.w;
  #pragma unroll
  for (int off = 16; off > 0; off >>= 1) sum += __shfl_down(sum, off, 32);
  if (lane == 0) out[tri] = sum;
}

// ---------------- host launcher ----------------

extern "C" void kernel_launch(void* const* d_in, const int* in_sizes, int n_in,
                              void* d_out, int out_size, void* d_ws, size_t ws_size,
                              hipStream_t stream) {
  const float* X        = (const float*)d_in[0];
  const float* W_rel    = (const float*)d_in[1];
  const float* W_self   = (const float*)d_in[2];
  const float* bvec     = (const float*)d_in[3];
  const float* rel_emb  = (const float*)d_in[4];
  const int*   edge_src = (const int*)d_in[5];
  const int*   edge_dst = (const int*)d_in[6];
  const int*   edge_rel = (const int*)d_in[7];
  const int*   tri_node = (const int*)d_in[8];
  const int*   tri_rel  = (const int*)d_in[9];
  float*       out      = (float*)d_out;

  unsigned int* wsu    = (unsigned int*)d_ws;
  unsigned int* cnt    = wsu + WS_CNT_U;
  unsigned int* hist   = wsu + WS_HIST_U;
  unsigned int* cursor = wsu + WS_CURSOR_U;
  unsigned int* offs   = wsu + WS_OFFS_U;
  int*          perm   = (int*)((char*)d_ws + WS_PERM_B);
  float*        agg    = (float*)((char*)d_ws + WS_AGG_B);

  k_zero<<<(WS_ZERO_U + 255) / 256, 256, 0, stream>>>(wsu);
  k_count<<<(E_EDGES + 255) / 256, 256, 0, stream>>>(edge_rel, edge_dst, cnt, hist);
  k_scan<<<1, 1, 0, stream>>>(hist, offs, cursor);
  k_bucket<<<(E_EDGES + 255) / 256, 256, 0, stream>>>(edge_rel, cursor, perm);
  k_self_gemm<<<(MTILES + 7) / 8, 256, 0, stream>>>(X, W_self, agg);
  {
    dim3 grid((E_EDGES + 127) / 128, R_REL);
    k_edge_gemm<<<grid, 256, 0, stream>>>(X, W_rel, edge_src, edge_dst,
                                          perm, offs, cnt, agg);
  }
  k_bias_relu<<<(N_NODES * EMB + 255) / 256, 256, 0, stream>>>(bvec, agg);
  k_score<<<(B_TRI * 32 + 255) / 256, 256, 0, stream>>>(agg, rel_emb, tri_node, tri_rel, out);
}